// GAT_8452495638848
// MI455X (gfx1250) — compile-verified
//
#include <hip/hip_runtime.h>
#include <math.h>

#define N 4096
#define NW (N / 32)          // 128 mask words per row
#define ALPHA 0.1f

typedef float v2f __attribute__((ext_vector_type(2)));
typedef float v8f __attribute__((ext_vector_type(8)));

__device__ __forceinline__ float lrelu(float x) { return x > 0.f ? x : ALPHA * x; }

// Pair-interleaved matrix layout: element (j, c) of an [Nrows][16] matrix is
// stored at (j>>1)*32 + c*2 + (j&1).  A lane needing rows {j0, j0+1} at one
// column c (the WMMA A/B fragment pattern, j0 even) loads ONE aligned b64.
__device__ __forceinline__ int pairIdx(int j, int c) {
    return ((j >> 1) << 5) + (c << 1) + (j & 1);
}

// ---------------------------------------------------------------------------
// Kernel 1: pack int32 adjacency into a bitmask (64MB -> 2MB, L2 resident)
// ---------------------------------------------------------------------------
__global__ void pack_mask(const int* __restrict__ adj, unsigned* __restrict__ bits) {
    int t = blockIdx.x * blockDim.x + threadIdx.x;   // one 32-col word
    if (t >= N * NW) return;
    const int* src = adj + (size_t)t * 32;
    unsigned w = 0;
#pragma unroll
    for (int b = 0; b < 32; ++b) w |= (src[b] > 0 ? 1u : 0u) << b;
    bits[t] = w;
}

// ---------------------------------------------------------------------------
// Kernel 2: zero-pad W [h][fin][dh] -> pair-interleaved Wp [h][fin][16]
// (removes all predication from the projection GEMM's B loads)
// ---------------------------------------------------------------------------
__global__ void pack_W(const float* __restrict__ W, float* __restrict__ Wp,
                       int heads, int fin, int dh) {
    int t = blockIdx.x * blockDim.x + threadIdx.x;
    if (t >= heads * fin * 16) return;
    int c  = t & 15;
    int kf = t >> 4;              // h*fin + k
    int k  = kf % fin;
    int h  = kf / fin;
    float v = (c < dh) ? W[(size_t)kf * dh + c] : 0.f;
    Wp[(size_t)h * fin * 16 + pairIdx(k, c)] = v;
}

// ---------------------------------------------------------------------------
// Kernel 3: projection GEMM  Wh[h] = x @ Wp[h]  via V_WMMA_F32_16X16X4_F32.
// One wave computes a 16x16 output tile; all loads unconditional; Wh stored
// pair-interleaved for the attention kernel's B-fragment loads.
// grid: (N/16, heads), block: 32
// ---------------------------------------------------------------------------
template <int FIN>
__global__ __launch_bounds__(32) void proj_wmma(const float* __restrict__ x,
                                                const float* __restrict__ Wp,
                                                float* __restrict__ Wh) {
    int h = blockIdx.y;
    int rowbase = (int)blockIdx.x * 16;
    int lane  = threadIdx.x & 31;
    int rc    = lane & 15;            // A-row / B-col / C-col
    int khalf = (lane >> 4) << 1;     // lanes 0-15 -> K{0,1}; 16-31 -> K{2,3}
    const float* Wt = Wp + h * (FIN * 16) + (rc << 1);   // + pair base for col rc
    const float* xr = x + (rowbase + rc) * FIN;
    v8f c = {};
#pragma unroll
    for (int k = 0; k < FIN; k += 4) {
        int ka = k + khalf;                        // even
        v2f a = *(const v2f*)(xr + ka);            // one b64: x[row][ka..ka+1]
        v2f b = *(const v2f*)(Wt + ((ka >> 1) << 5)); // one b64: Wp[ka..ka+1][rc]
        c = __builtin_amdgcn_wmma_f32_16x16x4_f32(false, a, false, b,
                                                  (short)0, c, false, false);
    }
    // C layout: VGPR r -> row r (lanes 0-15) / row r+8 (lanes 16-31), col = lane&15
    float* dst = Wh + h * (N * 16);
    int rhi = (lane >> 4) << 3;       // 0 or 8
#pragma unroll
    for (int r = 0; r < 8; ++r)
        dst[pairIdx(rowbase + rhi + r, rc)] = c[r];
}

// ---------------------------------------------------------------------------
// Kernel 4: s = Wh . a_src, d = Wh . a_dst  (per head, per node)
// ---------------------------------------------------------------------------
__global__ void attn_coeffs(const float* __restrict__ Wh,
                            const float* __restrict__ a_src,
                            const float* __restrict__ a_dst,
                            float* __restrict__ s, float* __restrict__ d,
                            int heads, int dh) {
    int t = blockIdx.x * blockDim.x + threadIdx.x;
    if (t >= heads * N) return;
    int h = t / N, n = t % N;
    const float* base = Wh + (size_t)h * N * 16;
    float ss = 0.f, dd = 0.f;
    for (int c = 0; c < dh; ++c) {
        float v = base[pairIdx(n, c)];
        ss += v * a_src[h * dh + c];
        dd += v * a_dst[h * dh + c];
    }
    s[t] = ss;
    d[t] = dd;
}

// ---------------------------------------------------------------------------
// Kernel 5: per-head global max of d (safe softmax shift; leaky_relu monotone)
// ---------------------------------------------------------------------------
__global__ void dmax_kernel(const float* __restrict__ d, float* __restrict__ dmax) {
    __shared__ float red[256];
    int h = blockIdx.x;
    float m = -1e30f;
    for (int i = threadIdx.x; i < N; i += 256) m = fmaxf(m, d[(size_t)h * N + i]);
    red[threadIdx.x] = m;
    __syncthreads();
    for (int st = 128; st > 0; st >>= 1) {
        if (threadIdx.x < st) red[threadIdx.x] = fmaxf(red[threadIdx.x], red[threadIdx.x + st]);
        __syncthreads();
    }
    if (threadIdx.x == 0) dmax[h] = red[0];
}

// ---------------------------------------------------------------------------
// Kernel 6: fused masked-softmax attention + aggregation (the hot loop).
// One wave owns 16 rows of one head; streams all 4096 columns in K=4 chunks,
// feeding exp-weights straight into V_WMMA_F32_16X16X4_F32 accumulators.
// Per chunk: 1 ds_load_b64 (d pair) + 1 global_load_b64 (Wh pair, L2-hot)
//            + ~12 VALU + 2 v_exp_f32 + 1 v_wmma.  No EXEC divergence.
// grid: (N/128, heads), block: 256 (8 waves)
// ---------------------------------------------------------------------------
template <int DH, int FOUT, int ELU>
__global__ __launch_bounds__(256) void att_aggregate(
    const unsigned* __restrict__ bits, const float* __restrict__ Wh,
    const float* __restrict__ s, const float* __restrict__ d,
    const float* __restrict__ dmax, float* __restrict__ out) {
    __shared__ float ldsD[N];                       // 16 KB of 320 KB WGP LDS
    int h    = blockIdx.y;
    int tid  = threadIdx.x;
    int lane = tid & 31;
    int wave = tid >> 5;
    for (int i = tid; i < N; i += 256) ldsD[i] = d[h * N + i];
    __syncthreads();

    int rowbase = (int)blockIdx.x * 128 + wave * 16;
    int rc  = lane & 15;                            // row (for A) / col (for B,C)
    int off = (lane >> 4) << 1;                     // K sub-offset: 0 or 2
    int row = rowbase + rc;
    float srow = s[h * N + row];
    float M    = lrelu(srow + dmax[h]);             // >= true row max
    const unsigned* mrow = bits + row * NW;
    const float* WhC = Wh + h * (N * 16) + (rc << 1) + (off << 4); // col+K-half base

    v8f   acc  = {};
    float lsum = 0.f;
    for (int jw = 0; jw < NW; ++jw) {
        unsigned mw = mrow[jw];
        const float* wj = WhC + (jw << 9);          // + jw*32 rows * 16
        const float* dj = ldsD + (jw << 5) + off;
#pragma unroll
        for (int cc = 0; cc < 8; ++cc) {            // 8 x (K=4) chunks per word
            int jb = (cc << 2) + off;               // bit position in mask word
            v2f dpair = *(const v2f*)(dj + (cc << 2));          // d[j0], d[j0+1]
            float e0 = lrelu(srow + dpair.x);
            float e1 = lrelu(srow + dpair.y);
            float p0 = ((mw >> jb)       & 1u) ? __expf(e0 - M) : 0.f;
            float p1 = ((mw >> (jb + 1)) & 1u) ? __expf(e1 - M) : 0.f;
            lsum += p0 + p1;
            v2f a; a.x = p0; a.y = p1;
            v2f b = *(const v2f*)(wj + (cc << 6));  // Wh[j0][rc], Wh[j0+1][rc]
            acc = __builtin_amdgcn_wmma_f32_16x16x4_f32(false, a, false, b,
                                                        (short)0, acc, false, false);
        }
    }
    // combine the two K-halves of the row sum (lane L and L+16 share a row)
    lsum += __shfl_xor(lsum, 16, 32);
    int hhi = (lane & 16) >> 1;                     // 0 or 8 (C VGPR row offset)
#pragma unroll
    for (int r = 0; r < 8; ++r) {
        float lr = __shfl(lsum, hhi + r, 32);       // normalizer of row hhi+r
        float v  = acc[r] / lr;
        if (ELU) v = v > 0.f ? v : __expf(v) - 1.f;
        if (rc < DH)
            out[(rowbase + hhi + r) * FOUT + h * DH + rc] = v;
    }
}

// ---------------------------------------------------------------------------
// Kernel 7: mean-pool over nodes + final FC
// ---------------------------------------------------------------------------
__global__ void pool_fc(const float* __restrict__ x5, const float* __restrict__ fc_w,
                        const float* __restrict__ fc_b, float* __restrict__ out) {
    __shared__ float red[256 * 8];
    float acc[8] = {};
    for (int i = threadIdx.x; i < N; i += 256) {
#pragma unroll
        for (int c = 0; c < 8; ++c) acc[c] += x5[i * 8 + c];
    }
#pragma unroll
    for (int c = 0; c < 8; ++c) red[threadIdx.x * 8 + c] = acc[c];
    __syncthreads();
    for (int st = 128; st > 0; st >>= 1) {
        if (threadIdx.x < st)
            for (int c = 0; c < 8; ++c)
                red[threadIdx.x * 8 + c] += red[(threadIdx.x + st) * 8 + c];
        __syncthreads();
    }
    if (threadIdx.x == 0) {
        float r = 0.f;
        for (int c = 0; c < 8; ++c) r += (red[c] / (float)N) * fc_w[c];
        out[0] = r + fc_b[0];
    }
}

// ---------------------------------------------------------------------------
extern "C" void kernel_launch(void* const* d_in, const int* in_sizes, int n_in,
                              void* d_out, int out_size, void* d_ws, size_t ws_size,
                              hipStream_t stream) {
    const float* x0  = (const float*)d_in[0];
    const int*   adj = (const int*)d_in[1];
    const float* W[5]  = {(const float*)d_in[2],  (const float*)d_in[5],
                          (const float*)d_in[8],  (const float*)d_in[11],
                          (const float*)d_in[14]};
    const float* As[5] = {(const float*)d_in[3],  (const float*)d_in[6],
                          (const float*)d_in[9],  (const float*)d_in[12],
                          (const float*)d_in[15]};
    const float* Ad[5] = {(const float*)d_in[4],  (const float*)d_in[7],
                          (const float*)d_in[10], (const float*)d_in[13],
                          (const float*)d_in[16]};
    const float* fcw = (const float*)d_in[17];
    const float* fcb = (const float*)d_in[18];
    float* out = (float*)d_out;

    // workspace layout (~8.8 MB)
    char* ws = (char*)d_ws;
    unsigned* bitsb = (unsigned*)ws;  ws += (size_t)N * NW * 4;      // 2 MB
    float* Wh    = (float*)ws;        ws += (size_t)8 * N * 16 * 4;  // 2 MB
    float* Wp    = (float*)ws;        ws += (size_t)8 * 256 * 16 * 4;// 128 KB
    float* sbuf  = (float*)ws;        ws += (size_t)8 * N * 4;       // 128 KB
    float* dbuf  = (float*)ws;        ws += (size_t)8 * N * 4;       // 128 KB
    float* dmaxb = (float*)ws;        ws += 256;
    float* bufA  = (float*)ws;        ws += (size_t)N * 128 * 4;     // 2 MB
    float* bufB  = (float*)ws;        ws += (size_t)N * 128 * 4;     // 2 MB

    pack_mask<<<(N * NW + 255) / 256, 256, 0, stream>>>(adj, bitsb);

    // ---- Layer 1: fin=256 heads=8 dh=16 fout=128 elu ----
    pack_W<<<(8 * 256 * 16 + 255) / 256, 256, 0, stream>>>(W[0], Wp, 8, 256, 16);
    proj_wmma<256><<<dim3(N / 16, 8), 32, 0, stream>>>(x0, Wp, Wh);
    attn_coeffs<<<(8 * N + 255) / 256, 256, 0, stream>>>(Wh, As[0], Ad[0], sbuf, dbuf, 8, 16);
    dmax_kernel<<<8, 256, 0, stream>>>(dbuf, dmaxb);
    att_aggregate<16, 128, 1><<<dim3(N / 128, 8), 256, 0, stream>>>(bitsb, Wh, sbuf, dbuf, dmaxb, bufA);

    // ---- Layer 2: fin=128 heads=8 dh=8 fout=64 elu ----
    pack_W<<<(8 * 128 * 16 + 255) / 256, 256, 0, stream>>>(W[1], Wp, 8, 128, 8);
    proj_wmma<128><<<dim3(N / 16, 8), 32, 0, stream>>>(bufA, Wp, Wh);
    attn_coeffs<<<(8 * N + 255) / 256, 256, 0, stream>>>(Wh, As[1], Ad[1], sbuf, dbuf, 8, 8);
    dmax_kernel<<<8, 256, 0, stream>>>(dbuf, dmaxb);
    att_aggregate<8, 64, 1><<<dim3(N / 128, 8), 256, 0, stream>>>(bitsb, Wh, sbuf, dbuf, dmaxb, bufB);

    // ---- Layer 3: fin=64 heads=4 dh=8 fout=32 elu ----
    pack_W<<<(4 * 64 * 16 + 255) / 256, 256, 0, stream>>>(W[2], Wp, 4, 64, 8);
    proj_wmma<64><<<dim3(N / 16, 4), 32, 0, stream>>>(bufB, Wp, Wh);
    attn_coeffs<<<(4 * N + 255) / 256, 256, 0, stream>>>(Wh, As[2], Ad[2], sbuf, dbuf, 4, 8);
    dmax_kernel<<<4, 256, 0, stream>>>(dbuf, dmaxb);
    att_aggregate<8, 32, 1><<<dim3(N / 128, 4), 256, 0, stream>>>(bitsb, Wh, sbuf, dbuf, dmaxb, bufA);

    // ---- Layer 4: fin=32 heads=1 dh=16 fout=16 no-elu ----
    pack_W<<<(1 * 32 * 16 + 255) / 256, 256, 0, stream>>>(W[3], Wp, 1, 32, 16);
    proj_wmma<32><<<dim3(N / 16, 1), 32, 0, stream>>>(bufA, Wp, Wh);
    attn_coeffs<<<(1 * N + 255) / 256, 256, 0, stream>>>(Wh, As[3], Ad[3], sbuf, dbuf, 1, 16);
    dmax_kernel<<<1, 256, 0, stream>>>(dbuf, dmaxb);
    att_aggregate<16, 16, 0><<<dim3(N / 128, 1), 256, 0, stream>>>(bitsb, Wh, sbuf, dbuf, dmaxb, bufB);

    // ---- Layer 5: fin=16 heads=1 dh=8 fout=8 no-elu (concat=False, h=1) ----
    pack_W<<<(1 * 16 * 16 + 255) / 256, 256, 0, stream>>>(W[4], Wp, 1, 16, 8);
    proj_wmma<16><<<dim3(N / 16, 1), 32, 0, stream>>>(bufB, Wp, Wh);
    attn_coeffs<<<(1 * N + 255) / 256, 256, 0, stream>>>(Wh, As[4], Ad[4], sbuf, dbuf, 1, 8);
    dmax_kernel<<<1, 256, 0, stream>>>(dbuf, dmaxb);
    att_aggregate<8, 8, 0><<<dim3(N / 128, 1), 256, 0, stream>>>(bitsb, Wh, sbuf, dbuf, dmaxb, bufA);

    pool_fc<<<1, 256, 0, stream>>>(bufA, fcw, fcb, out);
}